// MeanMaxStdPooling_36825049596586
// MI455X (gfx1250) — compile-verified
//
#include <hip/hip_runtime.h>
#include <float.h>
#include <math.h>

typedef __attribute__((ext_vector_type(2))) float v2f;
typedef __attribute__((ext_vector_type(4))) float v4f;
typedef __attribute__((ext_vector_type(8))) float v8f;

#define LN_EPS 1e-5f
#define LEAKY_SLOPE 0.01f
#define TK 64  // K-chunk staged in LDS for the WMMA GEMM

// ---------------------------------------------------------------------------
// Phase 0a: init segment-start offsets to N (handles trailing empty segments
// and starts[S] == N sentinel).
// ---------------------------------------------------------------------------
__global__ void init_starts_kernel(int* __restrict__ starts, int S, int N) {
    int i = blockIdx.x * blockDim.x + threadIdx.x;
    if (i <= S) starts[i] = N;
}

// ---------------------------------------------------------------------------
// Phase 0b: batch is sorted, so starts[s] = first row index i with batch[i]>=s.
// Each boundary is written by exactly one thread (no races).
// ---------------------------------------------------------------------------
__global__ void seg_starts_kernel(const int* __restrict__ batch,
                                  int* __restrict__ starts, int N) {
    int i = blockIdx.x * blockDim.x + threadIdx.x;
    if (i >= N) return;
    int b0 = batch[i];
    int bp = (i == 0) ? -1 : batch[i - 1];
    for (int s = bp + 1; s <= b0; ++s) starts[s] = i;
}

// ---------------------------------------------------------------------------
// Phase 1: segment pooling (mean | max | std). One block per segment,
// 128 threads x float4 columns (d=512). x is streamed exactly once with
// non-temporal b128 loads (410 MB > 192 MB L2: keep L2 for W/pooled/h).
// pooled layout: [S, 3d] = [mean(0..d) | max(d..2d) | std(2d..3d)]
// ---------------------------------------------------------------------------
__global__ void pool_kernel(const float* __restrict__ x,
                            const int* __restrict__ starts,
                            float* __restrict__ pooled, int d, int K) {
    int seg = blockIdx.x;
    int col = threadIdx.x * 4;          // first of 4 columns owned by thread
    int r0 = starts[seg], r1 = starts[seg + 1];
    int stride4 = d >> 2;               // row stride in v4f units

    v4f s  = {0.f, 0.f, 0.f, 0.f};
    v4f q  = {0.f, 0.f, 0.f, 0.f};
    v4f mx = {-FLT_MAX, -FLT_MAX, -FLT_MAX, -FLT_MAX};

    const v4f* p = (const v4f*)(x + (size_t)r0 * d + col);
    int rows = r1 - r0;
#pragma unroll 4
    for (int r = 0; r < rows; ++r, p += stride4) {
        v4f v = __builtin_nontemporal_load(p);   // global_load_b128, TH=NT
        s += v;
        q += v * v;
        mx.x = fmaxf(mx.x, v.x);
        mx.y = fmaxf(mx.y, v.y);
        mx.z = fmaxf(mx.z, v.z);
        mx.w = fmaxf(mx.w, v.w);
    }

    float count = (float)rows;
    float cnt   = fmaxf(count, 1.f);
    float rdiv  = 1.f / fmaxf(count - 1.f, 1.f);  // unbiased divisor clamp

    v4f mean, sd;
    float* out = pooled + (size_t)seg * K;
#pragma unroll
    for (int j = 0; j < 4; ++j) {
        float mj = s[j] / cnt;
        float var = (q[j] - mj * s[j]) * rdiv;    // torch_scatter std semantics
        var = fmaxf(var, 0.f);
        mean[j] = mj;
        sd[j]   = (var > 0.f) ? sqrtf(var) : 0.f;
    }
    *(v4f*)(out + col)         = mean;
    *(v4f*)(out + d + col)     = mx;   // empty segment -> -FLT_MAX == finfo.min
    *(v4f*)(out + 2 * d + col) = sd;
}

// ---------------------------------------------------------------------------
// Phase 2: H = pooled @ W + b using V_WMMA_F32_16X16X4_F32 (full f32 WMMA).
// grid = (M/16, Ncols/128), block = 256 threads = 8 waves.
// Each wave owns one 16x16 C tile; A 16xTK chunk staged in LDS and shared by
// all 8 waves; B (W) loaded from global (L2-resident, reused by all M-tiles).
//
// ISA layouts (7.12.2, wave32):
//   A 16x4 f32:  lane l<16 -> row M=l, v0=K0, v1=K1 ; lane l>=16 -> M=l-16,
//                v0=K2, v1=K3.
//   B 4x16 f32:  lanes 0-15 hold rows K0(v0),K1(v1); lanes 16-31 K2(v0),K3(v1)
//                at column N = lane&15.
//   C/D v8f:     element r -> M = r + (lane<16 ? 0 : 8), N = lane&15.
// ---------------------------------------------------------------------------
__global__ void gemm_wmma_kernel(const float* __restrict__ A,
                                 const float* __restrict__ W,
                                 const float* __restrict__ bias,
                                 float* __restrict__ H,
                                 int M, int N, int K) {
    __shared__ float As[16 * TK];  // 4 KB

    int wave  = threadIdx.x >> 5;   // 0..7
    int lane  = threadIdx.x & 31;
    int mbase = blockIdx.x * 16;
    int n0    = blockIdx.y * 128 + wave * 16;
    int mrow  = lane & 15;
    int klo   = (lane < 16) ? 0 : 2;

    v8f acc = {};

    for (int kc = 0; kc < K; kc += TK) {
        __syncthreads();
        // cooperative A-chunk load: 16*TK floats, uniform trip count (no
        // divergence -> EXEC stays all-ones for the WMMAs below)
        for (int t = threadIdx.x; t < 16 * TK; t += 256) {
            int r = t >> 6;          // t / TK   (TK == 64)
            int c = t & (TK - 1);    // t % TK
            As[t] = A[(size_t)(mbase + r) * K + kc + c];
        }
        __syncthreads();

#pragma unroll
        for (int kk = 0; kk < TK; kk += 4) {
            v2f a, b;
            a.x = As[mrow * TK + kk + klo];
            a.y = As[mrow * TK + kk + klo + 1];
            const float* wp = W + (size_t)(kc + kk + klo) * N + n0 + mrow;
            b.x = wp[0];
            b.y = wp[N];
            // D = A*B + C, f32 in / f32 accumulate
            acc = __builtin_amdgcn_wmma_f32_16x16x4_f32(
                false, a, false, b, (short)0, acc, false, false);
        }
    }

    int   ncol = n0 + mrow;
    int   mtop = mbase + ((lane < 16) ? 0 : 8);
    float bv   = bias[ncol];
#pragma unroll
    for (int r = 0; r < 8; ++r)
        H[(size_t)(mtop + r) * N + ncol] = acc[r] + bv;
}

// ---------------------------------------------------------------------------
// Phase 3: LayerNorm (eps=1e-5, biased var) + affine + LeakyReLU(0.01).
// One block per row, one thread per column, wave32 shfl + LDS reduction.
// ---------------------------------------------------------------------------
__global__ void ln_leaky_kernel(const float* __restrict__ H,
                                const float* __restrict__ gamma,
                                const float* __restrict__ beta,
                                float* __restrict__ out, int d) {
    __shared__ float wsum[32];
    __shared__ float wsq[32];
    __shared__ float s_mean, s_rstd;

    int row = blockIdx.x;
    int col = threadIdx.x;

    float v = H[(size_t)row * d + col];
    float s = v;
    float q = v * v;
#pragma unroll
    for (int off = 16; off > 0; off >>= 1) {  // wave32 reduction
        s += __shfl_down(s, off);
        q += __shfl_down(q, off);
    }
    int wid  = threadIdx.x >> 5;
    int lane = threadIdx.x & 31;
    if (lane == 0) { wsum[wid] = s; wsq[wid] = q; }
    __syncthreads();
    if (threadIdx.x == 0) {
        int   nw = blockDim.x >> 5;
        float ts = 0.f, tq = 0.f;
        for (int i = 0; i < nw; ++i) { ts += wsum[i]; tq += wsq[i]; }
        float mean = ts / (float)d;
        float var  = tq / (float)d - mean * mean;
        s_mean = mean;
        s_rstd = rsqrtf(fmaxf(var, 0.f) + LN_EPS);
    }
    __syncthreads();

    float y = (v - s_mean) * s_rstd * gamma[col] + beta[col];
    out[(size_t)row * d + col] = (y >= 0.f) ? y : LEAKY_SLOPE * y;
}

// ---------------------------------------------------------------------------
// Host-side launcher. Inputs (setup_inputs order):
//   0:x [N,d] f32   1:W [3d,d] f32   2:b [d] f32   3:gamma [d]   4:beta [d]
//   5:batch [N] i32 (sorted)         6:num_segments (scalar)
// Output: [S, d] f32.  ws layout: starts[(S+1)] | pooled[S,3d] | h[S,d]
// ---------------------------------------------------------------------------
extern "C" void kernel_launch(void* const* d_in, const int* in_sizes, int n_in,
                              void* d_out, int out_size, void* d_ws, size_t ws_size,
                              hipStream_t stream) {
    const float* x     = (const float*)d_in[0];
    const float* W     = (const float*)d_in[1];
    const float* b     = (const float*)d_in[2];
    const float* gamma = (const float*)d_in[3];
    const float* beta  = (const float*)d_in[4];
    const int*   batch = (const int*)d_in[5];

    int d = in_sizes[2];          // 512
    int N = in_sizes[0] / d;      // 200000
    int S = out_size / d;         // 1024
    int K = 3 * d;                // 1536

    char*  ws = (char*)d_ws;
    size_t starts_bytes = (((size_t)(S + 1) * sizeof(int)) + 255) & ~(size_t)255;
    int*   starts = (int*)ws;
    float* pooled = (float*)(ws + starts_bytes);
    float* h      = pooled + (size_t)S * (size_t)K;

    init_starts_kernel<<<(S + 256) / 256, 256, 0, stream>>>(starts, S, N);
    seg_starts_kernel<<<(N + 255) / 256, 256, 0, stream>>>(batch, starts, N);

    pool_kernel<<<S, d / 4, 0, stream>>>(x, starts, pooled, d, K);

    dim3 gg(S / 16, d / 128);
    gemm_wmma_kernel<<<gg, 256, 0, stream>>>(pooled, W, b, h, S, d, K);

    ln_leaky_kernel<<<S, d, 0, stream>>>(h, gamma, beta, (float*)d_out, d);
}